// CoordsToNRF_6073083756911
// MI455X (gfx1250) — compile-verified
//
#include <hip/hip_runtime.h>
#include <math.h>

// CDNA5 / gfx1250: wave32, V_WMMA_F32_16X16X4_F32.
typedef __attribute__((ext_vector_type(2))) float v2f;
typedef __attribute__((ext_vector_type(8))) float v8f;

#define N_ATOMS 128
#define NC2     8128                                 // 128*127/2
#define NRF_SCALE (627.5095f * 0.529177f / 100.0f)   // AU2KCALMOLA / MAX_NRF

// Pack one LDS row (x,y,z,pad) into a 16x4-f32 A/B WMMA fragment:
// lanes 0-15 carry (K0,K1)=(x,y); lanes 16-31 carry (K2,K3)=(z,0).
__device__ __forceinline__ v2f make_frag(const float4 x, int half) {
    v2f f;
    f.x = half ? x.z : x.x;
    f.y = half ? 0.0f : x.y;
    return f;
}

__global__ __launch_bounds__(256)
void coords_to_nrf_kernel(const float* __restrict__ coords,   // [B, 128, 3]
                          const float* __restrict__ atoms,    // [8128]
                          float* __restrict__ out)            // [B, 8128]
{
    __shared__ float4 sX[N_ATOMS];   // scaled x, y, z, |x|^2
    __shared__ float  sN[N_ATOMS];   // scaled |x|^2, contiguous for row loads

    const int b   = blockIdx.x;
    const int tid = threadIdx.x;

    // Pre-scale coords by 1/sqrt(NRF_SCALE): then d2' = d2/NRF_SCALE and
    // val = atoms[p] * rcp(d2') -- removes one multiply per output element.
    // (Literal argument -> sqrtf constant-folds at compile time.)
    const float cs = sqrtf(1.0f / NRF_SCALE);

    if (tid < N_ATOMS) {
        const float* c = coords + ((size_t)b * N_ATOMS + tid) * 3;
        const float x = c[0] * cs, y = c[1] * cs, z = c[2] * cs;
        const float n = x * x + y * y + z * z;
        sX[tid] = make_float4(x, y, z, n);
        sN[tid] = n;
    }
    __syncthreads();

    const int wave = tid >> 5;
    const int lane = tid & 31;
    const int half = lane >> 4;      // 0: lanes 0-15, 1: lanes 16-31
    const int l16  = lane & 15;

    float* __restrict__ outb = out + (size_t)b * NC2;
    const float4* sN4 = (const float4*)sN;

    // ---- Diagonal tile (wave, wave): 8 tiles for 8 waves; strict i>j guard
    //      (also keeps p <= 8127). A and B fragments are identical. ----
    {
        const float4 xb = sX[wave * 16 + l16];
        const v2f a = make_frag(xb, half);

        v8f g = {};
        g = __builtin_amdgcn_wmma_f32_16x16x4_f32(
                false, a, false, a, (short)0, g, false, false);

        const int   i_base = wave * 16 + 8 * half;   // M = r + 8*half
        const int   j      = wave * 16 + l16;
        const float nj     = xb.w;
        const float4 nlo = sN4[i_base >> 2];
        const float4 nhi = sN4[(i_base >> 2) + 1];
        const float nv[8] = {nlo.x, nlo.y, nlo.z, nlo.w,
                             nhi.x, nhi.y, nhi.z, nhi.w};

        unsigned rb = (unsigned)(i_base * (i_base - 1)) >> 1;  // i*(i-1)/2
        #pragma unroll
        for (int r = 0; r < 8; ++r) {
            const int i = i_base + r;
            if (i > j) {
                const unsigned p  = rb + (unsigned)j;
                const float    d2 = nv[r] + nj - 2.0f * g[r];
                const float   val = atoms[p] * __builtin_amdgcn_rcpf(d2);
                __builtin_nontemporal_store(val, outb + p);
            }
            rb += (unsigned)i;      // i*(i-1)/2 -> (i+1)*i/2
        }
    }

    // ---- 28 strictly-lower tiles (ti > tj): branch-free ----
    for (int t = wave; t < 28; t += 8) {
        // Row-major decode over {(1,0),(2,0),(2,1),(3,0),...}: row ti has ti tiles.
        int ti = 1, acc = 0;
        while (acc + ti <= t) { acc += ti; ++ti; }
        const int tj = t - acc;

        const float4 xa = sX[ti * 16 + l16];
        const float4 xb = sX[tj * 16 + l16];
        const v2f a  = make_frag(xa, half);
        const v2f bv = make_frag(xb, half);

        v8f g = {};
        g = __builtin_amdgcn_wmma_f32_16x16x4_f32(
                false, a, false, bv, (short)0, g, false, false);

        const int   i_base = ti * 16 + 8 * half;
        const int   j      = tj * 16 + l16;
        const float nj     = xb.w;
        const float4 nlo = sN4[i_base >> 2];
        const float4 nhi = sN4[(i_base >> 2) + 1];
        const float nv[8] = {nlo.x, nlo.y, nlo.z, nlo.w,
                             nhi.x, nhi.y, nhi.z, nhi.w};

        // Phase 1: compute all pair indices and issue all 8 atom loads
        // (8-deep memory-level parallelism per wave).
        unsigned pidx[8];
        float    aval[8];
        unsigned rb = (unsigned)(i_base * (i_base - 1)) >> 1;
        #pragma unroll
        for (int r = 0; r < 8; ++r) {
            pidx[r] = rb + (unsigned)j;
            aval[r] = atoms[pidx[r]];
            rb += (unsigned)(i_base + r);
        }

        // Phase 2: finish the arithmetic and stream the stores.
        #pragma unroll
        for (int r = 0; r < 8; ++r) {
            const float d2  = nv[r] + nj - 2.0f * g[r];
            const float val = aval[r] * __builtin_amdgcn_rcpf(d2);
            __builtin_nontemporal_store(val, outb + pidx[r]);
        }
    }
}

extern "C" void kernel_launch(void* const* d_in, const int* in_sizes, int n_in,
                              void* d_out, int out_size, void* d_ws, size_t ws_size,
                              hipStream_t stream) {
    (void)in_sizes; (void)n_in; (void)d_ws; (void)ws_size;
    const float* coords = (const float*)d_in[0];   // [B, 128, 3] f32
    const float* atoms  = (const float*)d_in[1];   // [8128] f32
    float* out = (float*)d_out;                    // [B, 8128] f32

    const int batch = out_size / NC2;              // 2048
    coords_to_nrf_kernel<<<batch, 256, 0, stream>>>(coords, atoms, out);
}